// PostProcess_71322226917992
// MI455X (gfx1250) — compile-verified
//
#include <hip/hip_runtime.h>
#include <stdint.h>

// ---------------------------------------------------------------------------
// DETR EDL postprocess for MI455X (gfx1250).
// Bandwidth-bound: one 210MB streaming read of logits -> ~9us floor @23.3TB/s.
// No matmul structure -> WMMA inapplicable; CDNA5 path used instead:
//   GLOBAL_LOAD_ASYNC_TO_LDS_B128 double-buffered staging + s_wait_asynccnt.
// Row sums: 8 threads/row, register-resident softplus, fixed-order shfl tree
// (bitwise-deterministic across replays).
// ---------------------------------------------------------------------------

#define B_IMG   64
#define Q_ROWS  9000
#define C_CLS   91
#define K_OUT   100
#define SLICES  25              // 9000/25 = 360 rows/block; 360 % 4 == 0 -> every tile 16B-aligned
#define ROWS_PB 360
#define TILE_R  32              // 32*91 floats = 11648 B per tile (16B multiple)
#define FULLT   11              // 11 full tiles of 32 rows
#define TAIL_R  8               // + 1 tail tile of 8 rows
#define NTILES  12
#define CAND    4096
#define KEEP    128             // keep >100 so compaction threshold stays exact
#define NEG_BIG (-3.0e38f)
#define JMAX    12              // ceil(91/8) elements per thread-stripe

// d_out layout (floats): scores[64*100] | labels[64*100] | boxes[64*100*4] | uncertainty[64*9000]
#define OFF_SCORES 0
#define OFF_LABELS 6400
#define OFF_BOXES  12800
#define OFF_UNC    38400

// ---- CDNA5 async global->LDS (ASYNCcnt) -----------------------------------
__device__ __forceinline__ void async_b128(unsigned lds_addr, const void* g) {
  asm volatile("global_load_async_to_lds_b128 %0, %1, off"
               :: "v"(lds_addr), "v"((unsigned long long)(uintptr_t)g)
               : "memory");
}
__device__ __forceinline__ void wait_async_le3() { asm volatile("s_wait_asynccnt 3" ::: "memory"); }
__device__ __forceinline__ void wait_async_le1() { asm volatile("s_wait_asynccnt 1" ::: "memory"); }
__device__ __forceinline__ void wait_async_0()  { asm volatile("s_wait_asynccnt 0" ::: "memory"); }

// Issue one tile's async copies. rows is 32 (728 b128 lanes -> 3 uniform
// wave-instructions) or 8 (182 lanes -> 1). Addresses are clamped so every
// lane stays active: per-wave ASYNCcnt increments are exactly 3 (or 1),
// making the pipelined s_wait_asynccnt bounds exact.
__device__ __forceinline__ void issue_tile(const float* g, float* l, int tid, int rows) {
  unsigned lbase = (unsigned)(uintptr_t)l;
  unsigned lanes = (unsigned)(rows * C_CLS) / 4u;       // 16B chunks
  unsigned iters = (lanes + 255u) / 256u;               // 3 or 1 (uniform)
  for (unsigned i = 0; i < iters; ++i) {
    unsigned f = (unsigned)tid + i * 256u;
    if (f > lanes - 1u) f = lanes - 1u;                 // clamp: duplicate copy, benign
    async_b128(lbase + f * 16u, (const char*)g + (size_t)f * 16u);
  }
}

// ---- descending bitonic sort of (score,idx) pairs in LDS ------------------
template <int N>
__device__ __forceinline__ void bitonic_desc(float* s, unsigned* ix, int tid) {
  for (int k = 2; k <= N; k <<= 1) {
    for (int j = k >> 1; j > 0; j >>= 1) {
      for (int i = tid; i < N; i += 256) {
        int l = i ^ j;
        if (l > i) {
          float a = s[i], c = s[l];
          unsigned ia = ix[i], il = ix[l];
          bool gt = (a > c) || (a == c && ia < il);      // idx tie-break => deterministic
          if (((i & k) == 0) ? !gt : gt) {
            s[i] = c; s[l] = a; ix[i] = il; ix[l] = ia;
          }
        }
      }
      __syncthreads();
    }
  }
}

// ---------------------------------------------------------------------------
// Kernel 1: stream logits (async-to-LDS double buffer), softplus in registers,
// 8-thread/row sums, uncertainty, threshold-filtered candidates, per-slice top-100.
// ---------------------------------------------------------------------------
__global__ __launch_bounds__(256) void k_edl_topk(
    const float* __restrict__ logits, const float* __restrict__ obj,
    float* __restrict__ uncert_out,
    float* __restrict__ ws_scores, unsigned* __restrict__ ws_idx)
{
  __shared__ float    tile[2][TILE_R * C_CLS];
  __shared__ float    cs[CAND];
  __shared__ unsigned ci[CAND];
  __shared__ float    rowscale[TILE_R];
  __shared__ int      s_cnt;
  __shared__ float    s_thr;

  const int tid  = threadIdx.x;
  const int b    = blockIdx.x / SLICES;
  const int sl   = blockIdx.x % SLICES;
  const int row0 = sl * ROWS_PB;
  const int myrow = tid >> 3;                            // 8 threads per row
  const int g     = tid & 7;                             // stripe within row
  const float* imgL = logits + (size_t)b * Q_ROWS * C_CLS;

  if (tid == 0) { s_cnt = 0; s_thr = NEG_BIG; }

  // prefetch tile 0
  issue_tile(imgL + (size_t)row0 * C_CLS, &tile[0][0], tid, TILE_R);
  __syncthreads();

  float sp[JMAX];

  for (int t = 0; t < NTILES; ++t) {
    const int rows  = (t < FULLT) ? TILE_R : TAIL_R;
    const int trow0 = row0 + t * TILE_R;
    const float* buf = &tile[t & 1][0];

    // software pipeline: issue next tile, then wait for current
    if (t + 1 < NTILES) {
      const int nrows = (t + 1 < FULLT) ? TILE_R : TAIL_R;
      issue_tile(imgL + (size_t)(trow0 + TILE_R) * C_CLS, &tile[(t + 1) & 1][0], tid, nrows);
      if (nrows == TILE_R) wait_async_le3(); else wait_async_le1();
    } else {
      wait_async_0();
    }
    __syncthreads();

    // Fused phase: softplus into registers + striped partial row sum.
    // Invalid classes 81..90 -> sp = 0 (alpha = 1).
    if (myrow < rows) {
      const float* rp = buf + myrow * C_CLS;
      float acc = 0.f;
      #pragma unroll
      for (int j = 0; j < JMAX; ++j) {
        int c = g + (j << 3);
        float v = 0.f;
        if (c < C_CLS) {
          float x = rp[c];
          if (c < 81) v = fmaxf(x, 0.f) + __logf(1.f + __expf(-fabsf(x)));
        }
        sp[j] = v;
        acc += v;
      }
      // fixed-order width-8 reduction tree (deterministic)
      acc += __shfl_down(acc, 4, 8);
      acc += __shfl_down(acc, 2, 8);
      acc += __shfl_down(acc, 1, 8);
      if (g == 0) {
        float asum = acc + (float)C_CLS;                 // +1 per class (incl. invalid)
        int gr = trow0 + myrow;
        float o = obj[(size_t)b * Q_ROWS + gr];
        rowscale[myrow] = __expf(-o) / asum;
        uncert_out[(size_t)b * Q_ROWS + gr] = (float)C_CLS / asum;
      }
    }
    __syncthreads();

    // Compact candidate buffer if the worst-case tile append could overflow.
    const int nelem = rows * C_CLS;
    if (s_cnt + nelem > CAND) {
      for (int i = s_cnt + tid; i < CAND; i += 256) { cs[i] = NEG_BIG; ci[i] = 0xFFFFFFFFu; }
      __syncthreads();
      bitonic_desc<CAND>(cs, ci, tid);
      if (tid == 0) {
        s_thr = fmaxf(s_thr, cs[KEEP - 1]);              // exact: KEEP > K_OUT
        s_cnt = KEEP;
      }
      __syncthreads();
    }
    const float th = s_thr;

    // Phase B: prob = rowscale * (sp + 1) straight from registers.
    if (myrow < rows) {
      const float rs = rowscale[myrow];
      const unsigned gbase = (unsigned)((trow0 + myrow) * C_CLS);
      #pragma unroll
      for (int j = 0; j < JMAX; ++j) {
        int c = g + (j << 3);
        if (c < C_CLS) {
          float prob = rs * (sp[j] + 1.f);
          if (prob > th) {
            int p = atomicAdd(&s_cnt, 1);                // bounded by compaction above
            cs[p] = prob;
            ci[p] = gbase + (unsigned)c;
          }
        }
      }
    }
    __syncthreads();
  }

  // Final: sort remaining candidates, emit exact per-slice top-100.
  const int cnt = s_cnt;
  for (int i = cnt + tid; i < CAND; i += 256) { cs[i] = NEG_BIG; ci[i] = 0xFFFFFFFFu; }
  __syncthreads();
  bitonic_desc<CAND>(cs, ci, tid);
  if (tid < K_OUT) {
    size_t o = ((size_t)b * SLICES + sl) * K_OUT + tid;
    ws_scores[o] = cs[tid];
    ws_idx[o]    = ci[tid];
  }
}

// ---------------------------------------------------------------------------
// Kernel 2: per image, merge 25*100 = 2500 slice candidates -> global top-100,
// then write scores/labels and gathered, cxcywh->xyxy, scaled boxes.
// ---------------------------------------------------------------------------
__global__ __launch_bounds__(256) void k_merge_emit(
    const float* __restrict__ ws_scores, const unsigned* __restrict__ ws_idx,
    const float* __restrict__ pred_boxes, const float* __restrict__ tsz,
    float* __restrict__ out)
{
  __shared__ float    cs[CAND];
  __shared__ unsigned ci[CAND];
  const int tid = threadIdx.x;
  const int b   = blockIdx.x;
  const int NC  = SLICES * K_OUT;                        // 2500

  for (int i = tid; i < CAND; i += 256) {
    if (i < NC) { cs[i] = ws_scores[(size_t)b * NC + i]; ci[i] = ws_idx[(size_t)b * NC + i]; }
    else        { cs[i] = NEG_BIG;                       ci[i] = 0xFFFFFFFFu; }
  }
  __syncthreads();
  bitonic_desc<CAND>(cs, ci, tid);

  if (tid < K_OUT) {
    float    sc  = cs[tid];
    unsigned gi  = ci[tid];
    unsigned row = gi / C_CLS;
    unsigned lab = gi - row * C_CLS;

    out[OFF_SCORES + (size_t)b * K_OUT + tid] = sc;
    out[OFF_LABELS + (size_t)b * K_OUT + tid] = (float)lab;

    const float* pb = pred_boxes + ((size_t)b * Q_ROWS + row) * 4;
    float cx = pb[0], cy = pb[1], w = pb[2], h = pb[3];
    float ih = tsz[b * 2 + 0];                            // img_h = target_sizes[:,0]
    float iw = tsz[b * 2 + 1];                            // img_w = target_sizes[:,1]
    float* bo = out + OFF_BOXES + ((size_t)b * K_OUT + tid) * 4;
    bo[0] = (cx - 0.5f * w) * iw;
    bo[1] = (cy - 0.5f * h) * ih;
    bo[2] = (cx + 0.5f * w) * iw;
    bo[3] = (cy + 0.5f * h) * ih;
  }
}

extern "C" void kernel_launch(void* const* d_in, const int* in_sizes, int n_in,
                              void* d_out, int out_size, void* d_ws, size_t ws_size,
                              hipStream_t stream) {
  (void)in_sizes; (void)n_in; (void)out_size; (void)ws_size;
  const float* pred_logits = (const float*)d_in[0];   // (64, 9000, 91)
  const float* pred_obj    = (const float*)d_in[1];   // (64, 9000)
  const float* pred_boxes  = (const float*)d_in[2];   // (64, 9000, 4)
  const float* target_sz   = (const float*)d_in[3];   // (64, 2)
  float* out = (float*)d_out;

  // workspace: per-slice candidate lists (scores then indices)
  const size_t ncand = (size_t)B_IMG * SLICES * K_OUT;  // 160000
  float*    ws_scores = (float*)d_ws;
  unsigned* ws_idx    = (unsigned*)((char*)d_ws + ncand * sizeof(float));

  k_edl_topk<<<B_IMG * SLICES, 256, 0, stream>>>(
      pred_logits, pred_obj, out + OFF_UNC, ws_scores, ws_idx);
  k_merge_emit<<<B_IMG, 256, 0, stream>>>(
      ws_scores, ws_idx, pred_boxes, target_sz, out);
}